// AreaLoss_7069516169625
// MI455X (gfx1250) — compile-verified
//
#include <hip/hip_runtime.h>
#include <hip/hip_bf16.h>

typedef __attribute__((ext_vector_type(2))) float v2f;
typedef __attribute__((ext_vector_type(4))) float v4f;
typedef __attribute__((ext_vector_type(8))) float v8f;

#define BATCH 32
#define NCH   1000
#define HW    3136          // 56*56 = 49 * 64
#define TOPK_ 25
#define NSEL  22            // indices 3..24

// ---------------------------------------------------------------------------
// K1: per-batch top-25 indices of main_out row (softmax is monotone -> skip).
// One lane per batch row; '>' keeps the smaller index on ties, matching
// jax.lax.top_k (stable descending sort).
// ---------------------------------------------------------------------------
__global__ void topk_kernel(const float* __restrict__ main_out,
                            int* __restrict__ sel) {
    int b = threadIdx.x;
    if (b >= BATCH) return;
    const float* row = main_out + b * NCH;
    int chosen[TOPK_];
    for (int r = 0; r < TOPK_; ++r) {
        float best = -__builtin_inff();
        int   bi   = 0;
        for (int i = 0; i < NCH; ++i) {
            bool taken = false;
            for (int t = 0; t < r; ++t) {
                if (chosen[t] == i) { taken = true; break; }
            }
            if (!taken) {
                float v = row[i];
                if (v > best) { best = v; bi = i; }
            }
        }
        chosen[r] = bi;
    }
    for (int r = 3; r < TOPK_; ++r) sel[b * NSEL + (r - 3)] = chosen[r];
}

// ---------------------------------------------------------------------------
// K2: one wave per selected channel. Sum 3136 floats with chained
// V_WMMA_F32_16X16X4_F32 (A = ones 16x4):
//   sum(accumulator) == 16 * sum(B elements)   [layout-agnostic identity]
// 49 tiles of 64 floats = 24 iterations x (1 b128 load -> 2 WMMAs on
// independent accumulators) + 1 remainder tile (b64 load).
// Uniform trip counts + single wave => EXEC all-1s at every WMMA by
// construction. Streaming reads are non-temporal (read-once, spare L2).
// ---------------------------------------------------------------------------
__global__ void __launch_bounds__(32)
feat_sum_kernel(const float* __restrict__ features,
                const int* __restrict__ sel,
                float* __restrict__ featp) {
    const int blk  = blockIdx.x;        // 0..703 == b*22 + j
    const int b    = blk / NSEL;
    const int c    = sel[blk];
    const float* chan = features + ((size_t)b * NCH + (size_t)c) * (size_t)HW;

    const int lane = threadIdx.x;       // 0..31, one full wave

    v2f ones; ones[0] = 1.0f; ones[1] = 1.0f;            // A = 16x4 of 1.0
    v8f acc0 = {0.f,0.f,0.f,0.f,0.f,0.f,0.f,0.f};
    v8f acc1 = {0.f,0.f,0.f,0.f,0.f,0.f,0.f,0.f};

    for (int t = 0; t < 24; ++t) {                       // tiles 0..47
        const v4f* srcp = (const v4f*)(chan + t * 128 + lane * 4);
        v4f q = __builtin_nontemporal_load(srcp);        // 512B/wave, 2 tiles
        v2f b0; b0[0] = q[0]; b0[1] = q[1];
        v2f b1; b1[0] = q[2]; b1[1] = q[3];
        acc0 = __builtin_amdgcn_wmma_f32_16x16x4_f32(
                   false, ones, false, b0, (short)0, acc0, false, false);
        acc1 = __builtin_amdgcn_wmma_f32_16x16x4_f32(
                   false, ones, false, b1, (short)0, acc1, false, false);
    }
    {                                                    // remainder tile 48
        const v2f* srcp = (const v2f*)(chan + 3072 + lane * 2);
        v2f q = __builtin_nontemporal_load(srcp);
        acc0 = __builtin_amdgcn_wmma_f32_16x16x4_f32(
                   false, ones, false, q, (short)0, acc0, false, false);
    }

    float s = 0.0f;
    for (int k = 0; k < 8; ++k) s += acc0[k] + acc1[k];

    // wave32 butterfly reduce (fixed order -> deterministic)
    for (int m = 16; m >= 1; m >>= 1) s += __shfl_xor(s, m, 32);

    if (lane == 0) featp[blk] = s * (1.0f / 16.0f);      // exact /16
}

// ---------------------------------------------------------------------------
// K3: deterministic final reduce: sum(p) + sum(channel partials), /denom.
// ---------------------------------------------------------------------------
__global__ void __launch_bounds__(256)
finalize_kernel(const float* __restrict__ p, int np,
                const float* __restrict__ featp, int nf,
                float* __restrict__ out) {
    __shared__ float red[256];
    float s = 0.0f;
    for (int i = threadIdx.x; i < np; i += 256) s += p[i];
    for (int i = threadIdx.x; i < nf; i += 256) s += featp[i];
    red[threadIdx.x] = s;
    __syncthreads();
    for (int st = 128; st > 0; st >>= 1) {
        if (threadIdx.x < st) red[threadIdx.x] += red[threadIdx.x + st];
        __syncthreads();
    }
    if (threadIdx.x == 0) out[0] = red[0] / (float)(BATCH * HW);
}

// ---------------------------------------------------------------------------
extern "C" void kernel_launch(void* const* d_in, const int* in_sizes, int n_in,
                              void* d_out, int out_size, void* d_ws, size_t ws_size,
                              hipStream_t stream) {
    const float* p        = (const float*)d_in[0];   // (32,1,56,56)
    const float* main_out = (const float*)d_in[1];   // (32,1000)
    const float* features = (const float*)d_in[2];   // (32,1000,56,56)
    float* out = (float*)d_out;

    int*   sel   = (int*)d_ws;                                        // 704 ints
    float* featp = (float*)((char*)d_ws + ((BATCH * NSEL * sizeof(int)
                                            + 255) & ~(size_t)255));  // 704 floats

    topk_kernel<<<1, 32, 0, stream>>>(main_out, sel);
    feat_sum_kernel<<<BATCH * NSEL, 32, 0, stream>>>(features, sel, featp);
    finalize_kernel<<<1, 256, 0, stream>>>(p, in_sizes[0], featp, BATCH * NSEL, out);
}